// BERT_BiLSTM_CRF_30416958390305
// MI455X (gfx1250) — compile-verified
//
#include <hip/hip_runtime.h>

// CRF NLL for MI455X (gfx1250, wave32).
// Forward recursion in the scaled-linear domain: per step, a 16-sentence batch
// does D = P(16x9) x exp(trans)^T(9x16) as three chained V_WMMA_F32_16X16X4_F32
// (K=9 padded to 12), then multiplies exp(emissions). Emission loads are
// software-pipelined one step ahead via incremented 32-bit offsets (saddr +
// scaled voffset addressing); last step peeled. Renormalization amortized to
// every 4th step — alpha = acc + log P holds whether or not rows are
// normalized or frozen.

typedef __attribute__((ext_vector_type(2))) float v2f;   // f32 A/B tile: 2 VGPRs
typedef __attribute__((ext_vector_type(8))) float v8f;   // f32 C/D tile: 8 VGPRs

#define KTAGS 9
#define START_TAG 7
#define STOP_TAG 8
#define RESCALE 4

// Sum across each 16-lane half of the wave32 using single-instruction
// ds_swizzle (group-of-32: imm = xor_mask<<10 | and_mask 0x1f).
__device__ __forceinline__ float halfsum16(float v) {
    v += __int_as_float(__builtin_amdgcn_ds_swizzle(__float_as_int(v), 0x041f)); // SWAPX1
    v += __int_as_float(__builtin_amdgcn_ds_swizzle(__float_as_int(v), 0x081f)); // SWAPX2
    v += __int_as_float(__builtin_amdgcn_ds_swizzle(__float_as_int(v), 0x101f)); // SWAPX4
    v += __int_as_float(__builtin_amdgcn_ds_swizzle(__float_as_int(v), 0x201f)); // SWAPX8
    return v;
}

__global__ __launch_bounds__(32)
void crf_forward_wmma(const float* __restrict__ feats,
                      const float* __restrict__ trans,
                      const int*   __restrict__ lengths,
                      float*       __restrict__ fwd_out,
                      int B, int T) {
    // 16x16 f32 relayout tile (D layout -> A layout), padded row stride 17.
    __shared__ float tile[16 * 17];

    const int lane = threadIdx.x;     // 0..31
    const int n    = lane & 15;       // column (C/D layout) / row M (A layout)
    const int half = lane >> 4;       // C/D: 0 -> rows 0..7, 1 -> rows 8..15
    const int koff = half * 2;        // A/B vgpr K offset (K=0/1 vs K=2/3)
    const int b0   = blockIdx.x * 16; // 16 sentences per wave

    // Constant B tiles: Bc[k,n] = exp(trans[n, 4c+k]), zero padded past K=9.
    // Zero columns n>=9 / rows i>=9 guarantee D[., n>=9] == 0 forever.
    v2f Bt[3];
    #pragma unroll
    for (int c = 0; c < 3; ++c) {
        const int i0 = 4 * c + koff;
        float bx = 0.f, by = 0.f;
        if (n < KTAGS && i0     < KTAGS) bx = __expf(trans[n * KTAGS + i0]);
        if (n < KTAGS && i0 + 1 < KTAGS) by = __expf(trans[n * KTAGS + i0 + 1]);
        Bt[c].x = bx; Bt[c].y = by;
    }
    // Termination weights exp(trans[STOP, j]) (trans[STOP,STOP] = -1e4 -> 0).
    const float e2 = (n < KTAGS) ? __expf(trans[STOP_TAG * KTAGS + n]) : 0.f;

    // Per-row state in C/D layout.
    float P[8];       // scaled forward probabilities (cols 9..15 always 0)
    float acc[8];     // log-scale accumulator
    int   len_r[8];   // per-row sentence length (uniform within a 16-lane half)
    int   off_r[8];   // 32-bit element offset of NEXT step's emission
    const int jcl = (n < KTAGS) ? n : (KTAGS - 1);
    float fv[8];      // software-pipelined emissions for the current step
    #pragma unroll
    for (int r = 0; r < 8; ++r) {
        const int b   = b0 + r + half * 8;
        const int bcl = (b < B) ? b : (B - 1);
        const int base = bcl * T * KTAGS + jcl;
        P[r]     = (n == START_TAG) ? 1.0f : 0.0f;     // exp(alpha0)
        acc[r]   = 0.0f;
        len_r[r] = (b < B) ? lengths[b] : 0;
        fv[r]    = feats[base];                        // t = 0 emissions
        off_r[r] = base + KTAGS;                       // -> t = 1
    }

    auto step = [&](int t, bool preload) {
        // 0) issue next step's emission loads NOW (32-bit incremented offsets
        //    => saddr + scaled voffset); their waits land after the LDS
        //    roundtrip + 3 WMMAs + 8 exps of this step.
        float fnext[8];
        if (preload) {
            #pragma unroll
            for (int r = 0; r < 8; ++r) {
                fnext[r]  = feats[off_r[r]];
                off_r[r] += KTAGS;
            }
        }

        // 1) spill P (C/D layout) to LDS (same-wave LDS is in-order; only
        //    code-motion barriers needed, no counter drain).
        #pragma unroll
        for (int r = 0; r < 8; ++r)
            tile[(r + half * 8) * 17 + n] = P[r];
        __builtin_amdgcn_wave_barrier();

        // 2) load all three A tiles into distinct registers (back-to-back DS
        //    issue, progressive dscnt waits), then chain 3 WMMAs.
        v2f a0, a1, a2;
        a0.x = tile[n * 17 + 0 + koff];  a0.y = tile[n * 17 + 1 + koff];
        a1.x = tile[n * 17 + 4 + koff];  a1.y = tile[n * 17 + 5 + koff];
        a2.x = tile[n * 17 + 8 + koff];  a2.y = tile[n * 17 + 9 + koff];
        v8f d = {};
        d = __builtin_amdgcn_wmma_f32_16x16x4_f32(false, a0, false, Bt[0],
                                                  (short)0, d, false, false);
        d = __builtin_amdgcn_wmma_f32_16x16x4_f32(false, a1, false, Bt[1],
                                                  (short)0, d, false, false);
        d = __builtin_amdgcn_wmma_f32_16x16x4_f32(false, a2, false, Bt[2],
                                                  (short)0, d, false, false);
        __builtin_amdgcn_wave_barrier();

        // 3) emission factors first (8 independent TRANS ops pipeline), then
        //    branchless freeze past each row's length. No n<9 masking needed:
        //    d[r] is exactly 0 in cols 9..15 and factor is finite.
        float factor[8];
        #pragma unroll
        for (int r = 0; r < 8; ++r) factor[r] = __expf(fv[r]);
        #pragma unroll
        for (int r = 0; r < 8; ++r) {
            const float u = d[r] * factor[r];
            P[r] = (t < len_r[r]) ? u : P[r];
            if (preload) fv[r] = fnext[r];
        }

        // 4) amortized renormalization (safe for frozen rows: rescaling any
        //    row at any time preserves alpha = acc + log P).
        if ((t & (RESCALE - 1)) == (RESCALE - 1)) {
            #pragma unroll
            for (int r = 0; r < 8; ++r) {
                const float s = halfsum16(P[r]);
                P[r]   *= __builtin_amdgcn_rcpf(s);
                acc[r] += __logf(s);
            }
        }
    };

    for (int t = 0; t + 1 < T; ++t) step(t, true);   // hot loop
    step(T - 1, false);                              // peeled final step

    // fwd = acc + log( sum_j P[j] * exp(trans[STOP, j]) )
    #pragma unroll
    for (int r = 0; r < 8; ++r) {
        const int b = b0 + r + half * 8;
        const float s = halfsum16(P[r] * e2);
        const float v = acc[r] + __logf(s);
        if (n == r && b < B) fwd_out[b] = v;
    }
}

__global__ __launch_bounds__(256)
void crf_gold(const float* __restrict__ feats,
              const float* __restrict__ trans,
              const int*   __restrict__ tags,
              const int*   __restrict__ lengths,
              float*       __restrict__ gold_out,
              int B, int T) {
    const int b = blockIdx.x * blockDim.x + threadIdx.x;
    if (b >= B) return;
    const int*   tg = tags  + (size_t)b * T;
    const float* f  = feats + (size_t)b * T * KTAGS;
    const int len = lengths[b];
    int   prev = START_TAG;
    float sc   = 0.f;
    for (int t = 0; t < len; ++t) {
        const int cur = tg[t];
        sc  += trans[cur * KTAGS + prev] + f[t * KTAGS + cur];
        prev = cur;
    }
    sc += trans[STOP_TAG * KTAGS + prev];
    gold_out[b] = sc;
}

__global__ __launch_bounds__(256)
void crf_reduce(const float* __restrict__ fwd,
                const float* __restrict__ gold,
                float* __restrict__ out, int B) {
    __shared__ float sm[256];
    float s = 0.f;
    for (int i = threadIdx.x; i < B; i += 256) s += fwd[i] - gold[i];
    sm[threadIdx.x] = s;
    __syncthreads();
    for (int st = 128; st > 0; st >>= 1) {
        if (threadIdx.x < st) sm[threadIdx.x] += sm[threadIdx.x + st];
        __syncthreads();
    }
    if (threadIdx.x == 0) out[0] = sm[0] / (float)B;
}

extern "C" void kernel_launch(void* const* d_in, const int* in_sizes, int n_in,
                              void* d_out, int out_size, void* d_ws, size_t ws_size,
                              hipStream_t stream) {
    const float* feats   = (const float*)d_in[0];   // [B,T,9]
    const float* trans   = (const float*)d_in[1];   // [9,9]
    const int*   tags    = (const int*)d_in[2];     // [B,T]
    const int*   lengths = (const int*)d_in[3];     // [B]

    const int B = in_sizes[3];
    const int T = in_sizes[2] / B;

    float* fwd_ws  = (float*)d_ws;       // [B]
    float* gold_ws = fwd_ws + B;         // [B]
    float* out     = (float*)d_out;      // scalar

    crf_forward_wmma<<<(B + 15) / 16, 32, 0, stream>>>(feats, trans, lengths,
                                                       fwd_ws, B, T);
    crf_gold<<<(B + 255) / 256, 256, 0, stream>>>(feats, trans, tags, lengths,
                                                  gold_ws, B, T);
    crf_reduce<<<1, 256, 0, stream>>>(fwd_ws, gold_ws, out, B);
}